// gtsms_fast_27341761806805
// MI455X (gfx1250) — compile-verified
//
#include <hip/hip_runtime.h>

#define N_IDS     600001
#define N_IDS_PAD 600004   // multiple of 4 -> tail-free b128 init; t1 stays 16B-aligned

typedef __attribute__((ext_vector_type(4))) int   i32x4;
typedef __attribute__((ext_vector_type(4))) float f32x4;

// ---------------------------------------------------------------------------
// Pass 0: zero both max tables (int bit patterns; 0x0 == 0.0f).
// Padded to a multiple of 4 ints -> pure unconditional b128 stores, no tail.
// ---------------------------------------------------------------------------
__global__ __launch_bounds__(256) void sm_init_tables(int* __restrict__ t, int n4) {
    int tid = blockIdx.x * blockDim.x + threadIdx.x;   // one i32x4 per thread
    if (tid < n4) {
        *(i32x4*)(t + 4 * tid) = (i32x4)0;
    }
}

// ---------------------------------------------------------------------------
// Pass 1: scatter-max. feat >= 0 and tables init to 0, so int-ordered atomic
// max on the float bit pattern is exact -> one non-returning
// global_atomic_max_i32 per (element, table), no CAS loop.
// pred_pair kept RT (re-read in pass 2); reg_feat streamed NT (read once).
// 4 elements per lane: 2x b128 pair loads + 1x b128 feat load.
// ---------------------------------------------------------------------------
__global__ __launch_bounds__(256) void sm_scatter_max(const int* __restrict__ pp,
                                                      const float* __restrict__ feat,
                                                      int* __restrict__ t0,
                                                      int* __restrict__ t1,
                                                      int n) {
    int e = (blockIdx.x * blockDim.x + threadIdx.x) * 4;
    if (e >= n) return;                          // n is a multiple of 4

    i32x4 p01 = *(const i32x4*)(pp + 2 * e);     // (id0,id1) of e, e+1   (32B aligned)
    i32x4 p23 = *(const i32x4*)(pp + 2 * e + 4); // (id0,id1) of e+2, e+3
    f32x4 f   = __builtin_nontemporal_load((const f32x4*)(feat + e));

    atomicMax(t0 + p01.x, __float_as_int(f.x));
    atomicMax(t1 + p01.y, __float_as_int(f.x));
    atomicMax(t0 + p01.z, __float_as_int(f.y));
    atomicMax(t1 + p01.w, __float_as_int(f.y));
    atomicMax(t0 + p23.x, __float_as_int(f.z));
    atomicMax(t1 + p23.y, __float_as_int(f.z));
    atomicMax(t0 + p23.z, __float_as_int(f.w));
    atomicMax(t1 + p23.w, __float_as_int(f.w));
}

// ---------------------------------------------------------------------------
// Pass 2: gather from the (L2-resident, ~4.8MB) tables, multiply, NT store.
// ---------------------------------------------------------------------------
__global__ __launch_bounds__(256) void sm_gather_mul(const int* __restrict__ pp,
                                                     const int* __restrict__ t0,
                                                     const int* __restrict__ t1,
                                                     float* __restrict__ out,
                                                     int n) {
    int e = (blockIdx.x * blockDim.x + threadIdx.x) * 4;
    if (e >= n) return;

    i32x4 p01 = *(const i32x4*)(pp + 2 * e);
    i32x4 p23 = *(const i32x4*)(pp + 2 * e + 4);

    f32x4 r;
    r.x = __int_as_float(t0[p01.x]) * __int_as_float(t1[p01.y]);
    r.y = __int_as_float(t0[p01.z]) * __int_as_float(t1[p01.w]);
    r.z = __int_as_float(t0[p23.x]) * __int_as_float(t1[p23.y]);
    r.w = __int_as_float(t0[p23.z]) * __int_as_float(t1[p23.w]);

    __builtin_nontemporal_store(r, (f32x4*)(out + e));
}

// ---------------------------------------------------------------------------
// Launch: init -> scatter-max -> gather-mul, all on `stream`.
// d_ws layout: [ t0: N_IDS_PAD ints ][ t1: N_IDS_PAD ints ]  (~4.8 MB)
// ---------------------------------------------------------------------------
extern "C" void kernel_launch(void* const* d_in, const int* in_sizes, int n_in,
                              void* d_out, int out_size, void* d_ws, size_t ws_size,
                              hipStream_t stream) {
    const int*   pp   = (const int*)d_in[0];     // pred_pair, (n,2) int32 interleaved
    const float* feat = (const float*)d_in[1];   // reg_feat,  (n,)  float32
    float*       out  = (float*)d_out;

    int n = in_sizes[1];                         // 33554432 (multiple of 4)

    int* t0 = (int*)d_ws;
    int* t1 = t0 + N_IDS_PAD;

    // Pass 0: zero 2*N_IDS_PAD ints (tail-free b128 stores).
    {
        int n4      = (2 * N_IDS_PAD) / 4;       // i32x4 stores
        int nblocks = (n4 + 255) / 256;
        sm_init_tables<<<nblocks, 256, 0, stream>>>(t0, n4);
    }

    // Pass 1 & 2: 4 elements per thread, 256 threads (8 wave32) per block.
    int nthreads = (n + 3) / 4;
    int nblocks  = (nthreads + 255) / 256;
    sm_scatter_max<<<nblocks, 256, 0, stream>>>(pp, feat, t0, t1, n);
    sm_gather_mul<<<nblocks, 256, 0, stream>>>(pp, t0, t1, out, n);
}